// DegreePrediction_40535901340065
// MI455X (gfx1250) — compile-verified
//
#include <hip/hip_runtime.h>
#include <math.h>

typedef float v2f __attribute__((ext_vector_type(2)));
typedef float v8f __attribute__((ext_vector_type(8)));

#define NN 80
#define LDST 81          // LDS row stride (81 mod 64 = 17, conflict-free gathers)
#define PI_TOL 0.01f
#define PI_ITERS 100
#define BLK 160          // 5 waves: one 16-row WMMA panel per wave

__launch_bounds__(BLK)
__global__ void dp_power_iter_kernel(const float* __restrict__ x,
                                     const float* __restrict__ r_zeros,
                                     const float* __restrict__ r_const,
                                     const float* __restrict__ w_t,
                                     const float* __restrict__ w_r,
                                     float* __restrict__ ws)
{
    __shared__ float Mlds[NN * LDST];
    __shared__ float vv[NN];
    __shared__ float uu[NN];
    __shared__ float sc[2];

    const int b   = blockIdx.x;            // b = i*80 + j
    const int tid = threadIdx.x;
    const long base = (long)b * (NN * NN); // contiguous 80x80 slice

    // ---- Fused build of M[b] = w_r * r_zeros + r_const into LDS (coalesced b128 loads) ----
    #pragma unroll
    for (int t = 0; t < 10; ++t) {
        int idx4 = tid + t * BLK;          // 0..1599 float4 chunks
        int idx  = idx4 * 4;               // col stays multiple of 4, never crosses a row
        float4 a = ((const float4*)(w_r     + base))[idx4];
        float4 z = ((const float4*)(r_zeros + base))[idx4];
        float4 c = ((const float4*)(r_const + base))[idx4];
        int row = idx / NN, col = idx % NN;
        float* dst = &Mlds[row * LDST + col];
        dst[0] = fmaf(a.x, z.x, c.x);
        dst[1] = fmaf(a.y, z.y, c.y);
        dst[2] = fmaf(a.z, z.z, c.z);
        dst[3] = fmaf(a.w, z.w, c.w);
    }
    if (tid < NN) vv[tid] = 0.11180339887498948f;  // 1/sqrt(80)
    __syncthreads();

    const int lane = tid & 31;
    const int wv   = tid >> 5;     // wave id 0..4 -> row panel
    const int m    = lane & 15;    // A/C row within panel
    const int g    = lane >> 4;    // K half selector
    const int row0 = wv * 16;

    // ---- Preload loop-invariant A fragments: v_wmma_f32_16x16x4_f32 layout ----
    // A 16x4 f32: lanes 0-15 hold K=0,1 in vgpr0,1; lanes 16-31 hold K=2,3.
    v2f aF[20];
    {
        const float* rowp = &Mlds[(row0 + m) * LDST];
        #pragma unroll
        for (int s = 0; s < 20; ++s) {
            aF[s].x = rowp[4 * s + 2 * g];
            aF[s].y = rowp[4 * s + 2 * g + 1];
        }
    }

    // ---- matvec: uu = M * vv via 20 accumulating WMMA f32 16x16x4 per panel ----
    // B 4x16 f32: vgpr p, lane-half g -> K = 2*g + p; column value broadcast = vv[k].
    // D 16x16 f32: vgpr j, lane L -> row m = j + 8*(L>>4), col n = L&15; extract n==0.
    auto matvec = [&]() {
        v8f acc = {};
        #pragma unroll
        for (int s = 0; s < 20; ++s) {
            v2f bF;
            bF.x = vv[4 * s + 2 * g];
            bF.y = vv[4 * s + 2 * g + 1];
            acc = __builtin_amdgcn_wmma_f32_16x16x4_f32(
                false, aF[s], false, bF, (short)0, acc, false, false);
        }
        if (m == 0) {
            #pragma unroll
            for (int j = 0; j < 8; ++j) uu[row0 + 8 * g + j] = acc[j];
        }
    };

    matvec();                       // u0 = A v0
    __syncthreads();

    float ev = 0.f;
    if (tid == 0) {                 // ev0 = v0 . u0
        float s = 0.f;
        for (int k = 0; k < NN; ++k) s += vv[k] * uu[k];
        ev = s;
    }

    // one matvec per iteration: v=u/||u||; u=A v; ev=v.u; freeze at first convergence
    for (int it = 0; it < PI_ITERS; ++it) {
        if (tid == 0) {
            float n2 = 0.f;
            for (int k = 0; k < NN; ++k) n2 += uu[k] * uu[k];
            sc[0] = 1.0f / sqrtf(n2);
        }
        __syncthreads();
        if (tid < NN) vv[tid] = uu[tid] * sc[0];
        __syncthreads();
        matvec();
        __syncthreads();
        if (tid == 0) {
            float s = 0.f;
            for (int k = 0; k < NN; ++k) s += vv[k] * uu[k];
            sc[1] = (fabsf(ev - s) < PI_TOL) ? 1.0f : 0.0f;
            ev = s;
        }
        __syncthreads();
        if (sc[1] != 0.0f) break;   // block-uniform
    }

    // ---- per-b contribution: ws[b*80+k] = v[k] * (x*w_t*r_const[i,j,i,i]) / v[i] ----
    if (tid == 0) {
        int i = b / NN, j = b % NN;
        float rcd  = r_const[base + i * NN + i];     // element (k=i,l=i) of the slice
        float tval = x[i * NN + j] * w_t[i * NN + j] * rcd;
        sc[0] = tval / vv[i];
    }
    __syncthreads();
    if (tid < NN) ws[(long)b * NN + tid] = vv[tid] * sc[0];
}

// Deterministic fixed-order reduction: out[k] = sum_b ws[b*80+k]
__global__ __launch_bounds__(256)
void dp_reduce_kernel(const float* __restrict__ ws, float* __restrict__ out)
{
    __shared__ float red[256];
    const int k = blockIdx.x;
    float s = 0.f;
    for (int b = threadIdx.x; b < NN * NN; b += 256)
        s += ws[(long)b * NN + k];
    red[threadIdx.x] = s;
    __syncthreads();
    for (int off = 128; off > 0; off >>= 1) {
        if (threadIdx.x < off) red[threadIdx.x] += red[threadIdx.x + off];
        __syncthreads();
    }
    if (threadIdx.x == 0) out[k] = red[0];
}

extern "C" void kernel_launch(void* const* d_in, const int* in_sizes, int n_in,
                              void* d_out, int out_size, void* d_ws, size_t ws_size,
                              hipStream_t stream) {
    (void)in_sizes; (void)n_in; (void)out_size; (void)ws_size;
    const float* x  = (const float*)d_in[0];
    const float* rz = (const float*)d_in[1];
    const float* rc = (const float*)d_in[2];
    const float* wt = (const float*)d_in[3];
    const float* wr = (const float*)d_in[4];
    float* out = (float*)d_out;
    float* ws  = (float*)d_ws;        // needs 6400*80*4 = 2,048,000 bytes

    dp_power_iter_kernel<<<NN * NN, BLK, 0, stream>>>(x, rz, rc, wt, wr, ws);
    dp_reduce_kernel<<<NN, 256, 0, stream>>>(ws, out);
}